// SoftXOR_67542655697466
// MI455X (gfx1250) — compile-verified
//
#include <hip/hip_runtime.h>
#include <hip/hip_bf16.h>

typedef __attribute__((ext_vector_type(2))) float v2f;
typedef __attribute__((ext_vector_type(8))) float v8f;

#define WAVES_PER_BLOCK 8

// Hadamard entry H16[a][b] = (-1)^popcount(a&b)
__device__ __forceinline__ float hsign(int a, int b) {
  return (__popc(a & b) & 1) ? -1.0f : 1.0f;
}

__device__ __forceinline__ float wave_max(float v) {
#pragma unroll
  for (int o = 16; o; o >>= 1) v = fmaxf(v, __shfl_xor(v, o, 32));
  return v;
}
__device__ __forceinline__ float wave_sum(float v) {
#pragma unroll
  for (int o = 16; o; o >>= 1) v += __shfl_xor(v, o, 32);
  return v;
}

// C = H16 @ V   (V row-major 16x16 in LDS). 4 chained wmma_f32_16x16x4.
__device__ __forceinline__ v8f mmul_HV(const float* V, int lane) {
  const int h = lane >> 4;      // which half of the wave
  const int c = lane & 15;      // M for A, N for B
  v8f acc = {};
#pragma unroll
  for (int kc = 0; kc < 4; ++kc) {
    const int k0 = 4 * kc + 2 * h;
    v2f A, B;
    A.x = hsign(c, k0);
    A.y = hsign(c, k0 + 1);
    B.x = V[k0 * 16 + c];
    B.y = V[(k0 + 1) * 16 + c];
    acc = __builtin_amdgcn_wmma_f32_16x16x4_f32(false, A, false, B, (short)0,
                                                acc, false, false);
  }
  return acc;
}

// C = V @ H16   (V row-major 16x16 in LDS). 4 chained wmma_f32_16x16x4.
__device__ __forceinline__ v8f mmul_VH(const float* V, int lane) {
  const int h = lane >> 4;
  const int c = lane & 15;
  v8f acc = {};
#pragma unroll
  for (int kc = 0; kc < 4; ++kc) {
    const int k0 = 4 * kc + 2 * h;
    v2f A, B;
    A.x = V[c * 16 + k0];
    A.y = V[c * 16 + k0 + 1];
    B.x = hsign(k0, c);
    B.y = hsign(k0 + 1, c);
    acc = __builtin_amdgcn_wmma_f32_16x16x4_f32(false, A, false, B, (short)0,
                                                acc, false, false);
  }
  return acc;
}

// Scatter a C/D fragment (lane l, reg r -> [r + 8*(l/16)][l%16]) to row-major LDS.
__device__ __forceinline__ void frag_store(float* L, const v8f& acc, int lane) {
  const int h = lane >> 4;
  const int n = lane & 15;
#pragma unroll
  for (int r = 0; r < 8; ++r) L[(r + 8 * h) * 16 + n] = acc[r];
}

__global__ __launch_bounds__(256) void SoftXOR_kernel(
    const float* __restrict__ x, const float* __restrict__ y,
    float* __restrict__ out, int Btot) {
  __shared__ __align__(16) float smem[WAVES_PER_BLOCK * 3 * 256];

  const int lane = threadIdx.x & 31;
  const int wave = threadIdx.x >> 5;
  const int b = blockIdx.x * WAVES_PER_BLOCK + wave;
  if (b >= Btot) return;  // wave-uniform: EXEC stays all-1s for WMMA

  float* Vx = smem + wave * 3 * 256;  // px matrix / scratch reuse
  float* Vy = Vx + 256;               // py matrix
  float* Vt = Vy + 256;               // relayout scratch

  // ---- coalesced row load: 2x b128 per operand per lane ----
  const float4* xr = (const float4*)(x + (size_t)b * 256);
  const float4* yr = (const float4*)(y + (size_t)b * 256);
  float4 x0 = xr[lane], x1 = xr[lane + 32];
  float4 y0 = yr[lane], y1 = yr[lane + 32];

  // ---- softmax(x) -> Vx ----
  {
    float m = fmaxf(fmaxf(fmaxf(x0.x, x0.y), fmaxf(x0.z, x0.w)),
                    fmaxf(fmaxf(x1.x, x1.y), fmaxf(x1.z, x1.w)));
    m = wave_max(m);
    float4 e0 = {__expf(x0.x - m), __expf(x0.y - m), __expf(x0.z - m), __expf(x0.w - m)};
    float4 e1 = {__expf(x1.x - m), __expf(x1.y - m), __expf(x1.z - m), __expf(x1.w - m)};
    float s = wave_sum(e0.x + e0.y + e0.z + e0.w + e1.x + e1.y + e1.z + e1.w);
    float inv = 1.0f / s;
    float4 p0 = {e0.x * inv, e0.y * inv, e0.z * inv, e0.w * inv};
    float4 p1 = {e1.x * inv, e1.y * inv, e1.z * inv, e1.w * inv};
    ((float4*)Vx)[lane] = p0;
    ((float4*)Vx)[lane + 32] = p1;
  }

  // ---- softmax(y) -> Vy ----
  {
    float m = fmaxf(fmaxf(fmaxf(y0.x, y0.y), fmaxf(y0.z, y0.w)),
                    fmaxf(fmaxf(y1.x, y1.y), fmaxf(y1.z, y1.w)));
    m = wave_max(m);
    float4 e0 = {__expf(y0.x - m), __expf(y0.y - m), __expf(y0.z - m), __expf(y0.w - m)};
    float4 e1 = {__expf(y1.x - m), __expf(y1.y - m), __expf(y1.z - m), __expf(y1.w - m)};
    float s = wave_sum(e0.x + e0.y + e0.z + e0.w + e1.x + e1.y + e1.z + e1.w);
    float inv = 1.0f / s;
    float4 p0 = {e0.x * inv, e0.y * inv, e0.z * inv, e0.w * inv};
    float4 p1 = {e1.x * inv, e1.y * inv, e1.z * inv, e1.w * inv};
    ((float4*)Vy)[lane] = p0;
    ((float4*)Vy)[lane + 32] = p1;
  }

  // ---- forward WHT: Xt = H Px H, Yt = H Py H ----
  v8f t;
  t = mmul_HV(Vx, lane); frag_store(Vt, t, lane);
  t = mmul_VH(Vt, lane); frag_store(Vx, t, lane);
  t = mmul_HV(Vy, lane); frag_store(Vt, t, lane);
  t = mmul_VH(Vt, lane); frag_store(Vy, t, lane);

  // ---- pointwise product with 1/256 normalization ----
  {
    const float sc = 1.0f / 256.0f;
    float4 a0 = ((float4*)Vx)[lane], a1 = ((float4*)Vx)[lane + 32];
    float4 c0 = ((float4*)Vy)[lane], c1 = ((float4*)Vy)[lane + 32];
    float4 q0 = {a0.x * c0.x * sc, a0.y * c0.y * sc, a0.z * c0.z * sc, a0.w * c0.w * sc};
    float4 q1 = {a1.x * c1.x * sc, a1.y * c1.y * sc, a1.z * c1.z * sc, a1.w * c1.w * sc};
    ((float4*)Vx)[lane] = q0;
    ((float4*)Vx)[lane + 32] = q1;
  }

  // ---- inverse WHT: Pz = H (Xt .* Yt / 256) H ----
  t = mmul_HV(Vx, lane); frag_store(Vt, t, lane);
  t = mmul_VH(Vt, lane);

  // ---- log + store (C-frag layout -> flat index (r+8h)*16 + n) ----
  {
    const int h = lane >> 4;
    const int n = lane & 15;
    float* orow = out + (size_t)b * 256;
#pragma unroll
    for (int r = 0; r < 8; ++r) {
      float pz = fmaxf(t[r], 1e-30f);  // guard rounding-induced non-positives
      orow[(r + 8 * h) * 16 + n] = __logf(pz);
    }
  }
}

extern "C" void kernel_launch(void* const* d_in, const int* in_sizes, int n_in,
                              void* d_out, int out_size, void* d_ws, size_t ws_size,
                              hipStream_t stream) {
  const float* x = (const float*)d_in[0];
  const float* y = (const float*)d_in[1];
  float* out = (float*)d_out;
  const int B = in_sizes[0] / 256;
  const int blocks = (B + WAVES_PER_BLOCK - 1) / WAVES_PER_BLOCK;
  SoftXOR_kernel<<<blocks, 256, 0, stream>>>(x, y, out, B);
}